// EnergyModel_523986010189
// MI455X (gfx1250) — compile-verified
//
#include <hip/hip_runtime.h>

typedef __attribute__((ext_vector_type(2))) float v2f;
typedef __attribute__((ext_vector_type(8))) float v8f;

#define NBATCH 32
#define NATOM  256
#define CIN    64
#define CH     128
#define RREL   5
#define RC     640        // RREL * CH
#define XDIM   1024
#define KTOT   1280       // NATOM * RREL
#define KCH    320        // K-chunk staged in LDS for attention GEMM
#define NEGV   (-9e15f)

__device__ __forceinline__ float leaky(float v) { return v >= 0.f ? v : 0.2f * v; }

__device__ __forceinline__ v8f wmma4(v2f a, v2f b, v8f c) {
  // D = A(16x4 f32) * B(4x16 f32) + C(16x16 f32), full-wave matrix op
  return __builtin_amdgcn_wmma_f32_16x16x4_f32(false, a, false, b, (short)0, c,
                                               false, false);
}

// LDS byte offset of a __shared__ object: generic -> addrspace(3) -> int.
// Passing this value (derived from the pointer) into the asm also makes the
// shared object escape, so the "memory" clobber forces the compiler to treat
// the DMA as writing it (reads after the wait cannot be folded to undef).
__device__ __forceinline__ unsigned lds_offset(const void* p) {
  return (unsigned)(unsigned long long)(__attribute__((address_space(3))) const char*)p;
}

// ---------------------------------------------------------------------------
// C[M,N] = A[M,K] @ W[K,N] + bias[N],  N = RC = 640, K compile-time (64/128)
// grid (M/16, 5), block 256 (8 waves). The 16 A-rows are shared by all 8
// waves: stage them once per block via gfx1250 async DMA-to-LDS (ASYNCcnt),
// then each wave runs one 16x16 WMMA tile with B streamed from L2.
// ---------------------------------------------------------------------------
template <int K>
__global__ void __launch_bounds__(256)
k_gemm_bias(const float* __restrict__ A, const float* __restrict__ W,
            const float* __restrict__ bias, float* __restrict__ C)
{
  __shared__ float sA[16 * K];
  const int tid  = threadIdx.x;
  const int lane = tid & 31;
  const int wave = tid >> 5;
  const int half = lane >> 4;          // K-pair select within the wave
  const int l16  = lane & 15;
  const int row0 = blockIdx.x * 16;
  const int col0 = (blockIdx.y * 8 + wave) * 16;

  // ---- async DMA: A tile (16 x K floats) -> LDS, 16B per lane-op ----------
#pragma unroll
  for (int i = 0; i < (16 * K) / 1024; ++i) {      // 1 (K=64) or 2 (K=128)
    const int g    = tid + i * 256;                // float4 group index
    const int row  = (g << 2) / K;
    const int colk = (g << 2) % K;
    const float*   gptr = A + (size_t)(row0 + row) * K + colk;
    const unsigned ldst = lds_offset(&sA[g << 2]); // real LDS byte offset
    asm volatile("global_load_async_to_lds_b128 %0, %1, off"
                 :: "v"(ldst), "v"(gptr) : "memory");
  }
  asm volatile("s_wait_asynccnt 0x0" ::: "memory");
  __syncthreads();

  // ---- WMMA mainloop (fully unrolled, constant LDS offsets) ---------------
  v8f acc = {};
#pragma unroll
  for (int k = 0; k < K; k += 4) {
    v2f a, b;
    a.x = sA[l16 * K + k + 2 * half];                     // A: M=l16, K pair
    a.y = sA[l16 * K + k + 2 * half + 1];
    b.x = W[(size_t)(k + 2 * half) * RC + col0 + l16];    // B: N=l16, K pair
    b.y = W[(size_t)(k + 2 * half + 1) * RC + col0 + l16];
    acc = wmma4(a, b, acc);
  }
  const float bv = bias[col0 + l16];
#pragma unroll
  for (int i = 0; i < 8; ++i) {
    const int m = row0 + i + 8 * half;                    // C/D VGPR layout
    C[(size_t)m * RC + col0 + l16] = acc[i] + bv;
  }
}

// ---------------------------------------------------------------------------
// src/dst attention projections: one thread per (b,n,r), 128-wide dots.
// ---------------------------------------------------------------------------
__global__ void k_srcdst(const float* __restrict__ h, const float* __restrict__ a,
                         float* __restrict__ src, float* __restrict__ dst)
{
  const int idx = blockIdx.x * blockDim.x + threadIdx.x;    // (b*N+n)*R + r
  if (idx >= NBATCH * NATOM * RREL) return;
  const int r = idx % RREL;
  const float* hrow = h + (size_t)idx * CH;
  const float* as = a + r * (2 * CH);
  const float* ad = as + CH;
  float s = 0.f, d = 0.f;
#pragma unroll 4
  for (int c = 0; c < CH; ++c) { const float v = hrow[c]; s += v * as[c]; d += v * ad[c]; }
  src[idx] = s; dst[idx] = d;
}

// ---------------------------------------------------------------------------
// Masked softmax + attention aggregation.
// Block = one (b, 16-row tile). Pass A: ONLINE softmax stats (single pass
// over the L2-resident mask). Pass B: stage normalized probs in a 16xKCH LDS
// chunk and run the (16 x 1280) @ (1280 x 128) GEMM with WMMA; wave w makes
// output columns [16w, 16w+16); next chunk's B rows are prefetched.
// ---------------------------------------------------------------------------
__global__ void __launch_bounds__(256)
k_attn(const float* __restrict__ srcb, const float* __restrict__ dstb,
       const int* __restrict__ bonds, const float* __restrict__ h,
       float* __restrict__ out, int apply_act)
{
  __shared__ float Pch[16 * KCH];          // 20 KB probability chunk
  __shared__ float ssrc[16 * RREL];
  __shared__ float rmax[16], rinv[16];

  const int b    = blockIdx.x >> 4;
  const int it   = blockIdx.x & 15;
  const int tid  = threadIdx.x;
  const int lane = tid & 31;
  const int wave = tid >> 5;
  const int half = lane >> 4;
  const int l16  = lane & 15;

  const int i0 = it * 16;
  const float* dsb = dstb + (size_t)b * NATOM * RREL;                 // dst[b, j*R+r]
  const int*   bnd = bonds + ((size_t)b * NATOM + i0) * NATOM * RREL; // mask rows
  const float* Hb  = h + (size_t)b * KTOT * CH;                       // (1280,128)

  if (tid < 16 * RREL)
    ssrc[tid] = srcb[((size_t)b * NATOM + i0 + tid / RREL) * RREL + tid % RREL];
  __syncthreads();

  // ---- Pass A: online per-row max & sum(exp) in ONE sweep -----------------
  for (int row = wave; row < 16; row += 8) {
    const int* brow = bnd + (size_t)row * NATOM * RREL;
    float m = -__builtin_inff(), s = 0.f;
    for (int k = lane; k < KTOT; k += 32) {
      float v = leaky(ssrc[row * RREL + k % RREL] + dsb[k]);
      v = (brow[k] == 1) ? v : NEGV;
      const float mn = fmaxf(m, v);
      s = s * __expf(m - mn) + __expf(v - mn);   // exp(-inf)=0 on first iter
      m = mn;
    }
    for (int off = 16; off > 0; off >>= 1) {     // wave32 (m,s) combine
      const float mo = __shfl_xor(m, off, 32);
      const float so = __shfl_xor(s, off, 32);
      const float mn = fmaxf(m, mo);
      s = s * __expf(m - mn) + so * __expf(mo - mn);
      m = mn;
    }
    if (lane == 0) { rmax[row] = m; rinv[row] = 1.f / s; }
  }
  __syncthreads();

  // ---- Pass B: chunked probs -> LDS, WMMA aggregation over K = 1280 -------
  const int col0 = wave * 16;
  v8f acc = {};
  for (int kc = 0; kc < KTOT; kc += KCH) {
    // prefetch next chunk's B rows while we do the elementwise fill
    if (kc + KCH < KTOT) {
      for (int rr = tid; rr < KCH; rr += 256) {
        const float* prow = Hb + (size_t)(kc + KCH + rr) * CH;
        __builtin_prefetch(prow, 0, 0);
        __builtin_prefetch(prow + 64, 0, 0);
      }
    }
    for (int idx = tid; idx < 16 * KCH; idx += 256) {
      const int row = idx / KCH, kk = idx % KCH;
      const int k = kc + kk;
      float v = leaky(ssrc[row * RREL + k % RREL] + dsb[k]);
      v = (bnd[(size_t)row * NATOM * RREL + k] == 1) ? v : NEGV;
      Pch[row * KCH + kk] = __expf(v - rmax[row]) * rinv[row];
    }
    __syncthreads();
#pragma unroll 8
    for (int k4 = 0; k4 < KCH; k4 += 4) {
      const int kg = kc + k4 + 2 * half;
      v2f a, bb;
      a.x  = Pch[l16 * KCH + k4 + 2 * half];        // A from LDS (probs)
      a.y  = Pch[l16 * KCH + k4 + 2 * half + 1];
      bb.x = Hb[(size_t)kg * CH + col0 + l16];      // B from L2-resident h
      bb.y = Hb[(size_t)(kg + 1) * CH + col0 + l16];
      acc = wmma4(a, bb, acc);
    }
    __syncthreads();
  }

#pragma unroll
  for (int i = 0; i < 8; ++i) {
    const int m = i0 + i + 8 * half;
    float v = acc[i];
    if (apply_act) v = leaky(v);
    out[((size_t)b * NATOM + m) * CH + col0 + l16] = v;
  }
}

// ---------------------------------------------------------------------------
// Mean + max pooling over atoms: feats[b] = [mean_j h, max_j h]  (B x 256)
// ---------------------------------------------------------------------------
__global__ void k_pool(const float* __restrict__ hfin, float* __restrict__ feats)
{
  const int b = blockIdx.x, c = threadIdx.x;   // 128 threads
  float s = 0.f, m = -__builtin_inff();
  for (int j = 0; j < NATOM; ++j) {
    const float v = hfin[((size_t)b * NATOM + j) * CH + c];
    s += v; m = fmaxf(m, v);
  }
  feats[b * 256 + c]       = s * (1.f / NATOM);
  feats[b * 256 + 128 + c] = m;
}

// ---------------------------------------------------------------------------
// z1 = leaky(concat(x, feats) @ We1 + be1)   (32 x 1280) @ (1280 x 256), WMMA
// grid 4 x 256 threads: 32 wave-tiles of 16x16. K loop split at the concat
// boundary so both halves are branchless.
// ---------------------------------------------------------------------------
__global__ void __launch_bounds__(256)
k_mlp1(const float* __restrict__ x, const float* __restrict__ feats,
       const float* __restrict__ We1, const float* __restrict__ be1,
       float* __restrict__ z1)
{
  const int lane = threadIdx.x & 31;
  const int wave = threadIdx.x >> 5;
  const int wg   = blockIdx.x * 8 + wave;      // 0..31
  const int row0 = (wg >> 4) * 16;             // 0 or 16
  const int col0 = (wg & 15) * 16;
  const int half = lane >> 4, l16 = lane & 15;
  const int ar   = row0 + l16;

  v8f acc = {};
  for (int k = 0; k < XDIM; k += 4) {          // x part of the concat
    const int ka = k + 2 * half;
    v2f a, b;
    a.x = x[ar * XDIM + ka];
    a.y = x[ar * XDIM + ka + 1];
    b.x = We1[(size_t)ka * 256 + col0 + l16];
    b.y = We1[(size_t)(ka + 1) * 256 + col0 + l16];
    acc = wmma4(a, b, acc);
  }
#pragma unroll 8
  for (int k = 0; k < 256; k += 4) {           // feats part of the concat
    const int ka = k + 2 * half;
    v2f a, b;
    a.x = feats[ar * 256 + ka];
    a.y = feats[ar * 256 + ka + 1];
    b.x = We1[(size_t)(XDIM + ka) * 256 + col0 + l16];
    b.y = We1[(size_t)(XDIM + ka + 1) * 256 + col0 + l16];
    acc = wmma4(a, b, acc);
  }
  const float bv = be1[col0 + l16];
#pragma unroll
  for (int i = 0; i < 8; ++i) {
    const int m = row0 + i + 8 * half;
    z1[m * 256 + col0 + l16] = leaky(acc[i] + bv);
  }
}

// ---------------------------------------------------------------------------
// z2 = leaky(z1 @ We2 + be2); out = z2 @ We3 + be3   (tiny, VALU)
// ---------------------------------------------------------------------------
__global__ void k_mlp23(const float* __restrict__ z1, const float* __restrict__ We2,
                        const float* __restrict__ be2, const float* __restrict__ We3,
                        const float* __restrict__ be3, float* __restrict__ out)
{
  __shared__ float z2[32 * 32];
  const int tid = threadIdx.x;
  for (int idx = tid; idx < 32 * 32; idx += 256) {
    const int row = idx >> 5, col = idx & 31;
    float s = be2[col];
    for (int k = 0; k < 256; ++k) s += z1[row * 256 + k] * We2[k * 32 + col];
    z2[idx] = leaky(s);
  }
  __syncthreads();
  if (tid < 32) {
    float s = be3[0];
    for (int k = 0; k < 32; ++k) s += z2[tid * 32 + k] * We3[k];
    out[tid] = s;
  }
}

// ---------------------------------------------------------------------------
extern "C" void kernel_launch(void* const* d_in, const int* in_sizes, int n_in,
                              void* d_out, int out_size, void* d_ws, size_t ws_size,
                              hipStream_t stream)
{
  (void)in_sizes; (void)n_in; (void)out_size; (void)ws_size;

  const float* x     = (const float*)d_in[0];
  const float* atoms = (const float*)d_in[1];
  const int*   bonds = (const int*)  d_in[2];
  const float* W1 = (const float*)d_in[3];  const float* b1 = (const float*)d_in[4];
  const float* a1 = (const float*)d_in[5];
  const float* W2 = (const float*)d_in[6];  const float* b2 = (const float*)d_in[7];
  const float* a2 = (const float*)d_in[8];
  const float* W3 = (const float*)d_in[9];  const float* b3 = (const float*)d_in[10];
  const float* a3 = (const float*)d_in[11];
  const float* We1 = (const float*)d_in[12]; const float* be1 = (const float*)d_in[13];
  const float* We2 = (const float*)d_in[14]; const float* be2 = (const float*)d_in[15];
  const float* We3 = (const float*)d_in[16]; const float* be3 = (const float*)d_in[17];
  float* outp = (float*)d_out;

  char* ws = (char*)d_ws;
  float* h     = (float*)ws; ws += (size_t)8192 * 640 * sizeof(float);   // 21 MB
  float* act   = (float*)ws; ws += (size_t)8192 * 128 * sizeof(float);   // 4 MB
  float* srcb  = (float*)ws; ws += (size_t)NBATCH * NATOM * RREL * sizeof(float);
  float* dstb  = (float*)ws; ws += (size_t)NBATCH * NATOM * RREL * sizeof(float);
  float* feats = (float*)ws; ws += (size_t)32 * 256 * sizeof(float);
  float* z1    = (float*)ws; ws += (size_t)32 * 256 * sizeof(float);

  const dim3 gg(512, 5), bb(256);

  // layer 1 (K = 64)
  k_gemm_bias<CIN><<<gg, bb, 0, stream>>>(atoms, W1, b1, h);
  k_srcdst<<<160, 256, 0, stream>>>(h, a1, srcb, dstb);
  k_attn<<<512, 256, 0, stream>>>(srcb, dstb, bonds, h, act, 1);
  // layer 2 (K = 128)
  k_gemm_bias<CH><<<gg, bb, 0, stream>>>(act, W2, b2, h);
  k_srcdst<<<160, 256, 0, stream>>>(h, a2, srcb, dstb);
  k_attn<<<512, 256, 0, stream>>>(srcb, dstb, bonds, h, act, 1);
  // layer 3 (no trailing activation)
  k_gemm_bias<CH><<<gg, bb, 0, stream>>>(act, W3, b3, h);
  k_srcdst<<<160, 256, 0, stream>>>(h, a3, srcb, dstb);
  k_attn<<<512, 256, 0, stream>>>(srcb, dstb, bonds, h, act, 0);
  // head
  k_pool<<<32, 128, 0, stream>>>(act, feats);
  k_mlp1<<<4, 256, 0, stream>>>(x, feats, We1, be1, z1);
  k_mlp23<<<1, 256, 0, stream>>>(z1, We2, be2, We3, be3, outp);
}